// Norm1d_80573586473071
// MI455X (gfx1250) — compile-verified
//
#include <hip/hip_runtime.h>
#include <stdint.h>

#define AFWD 0.999f
#define EPS_ 1e-5f

constexpr int CHUNK  = 128;          // rows per batch-chunk (a^128 ~ 0.88, numerically safe)
constexpr int TILE   = 32;           // rows per LDS tile
constexpr int NTILES = CHUNK / TILE; // 4
constexpr int BF_BLK = 32;           // features per (wave32) block
static_assert(CHUNK == 128, "aL computed as AFWD^128 via 7 squarings");

// ---- CDNA5 async-copy primitives (inline asm; portable across toolchains) ----
__device__ __forceinline__ void wait_async_le8() { asm volatile("s_wait_asynccnt 0x8" ::: "memory"); }
__device__ __forceinline__ void wait_async_le0() { asm volatile("s_wait_asynccnt 0x0" ::: "memory"); }
__device__ __forceinline__ void wait_ds_le0()    { asm volatile("s_wait_dscnt 0x0"    ::: "memory"); }

// Issue one 32-row x 32-feature tile (4KB) as 8 async b128 copies.
// lane layout: 8 lanes per row (16B each), 4 rows per issue.
// LAST_USE=true adds th:TH_LOAD_LU (non-temporal + discard on hit): pass 3 is
// the final consumer of x, so free L2 capacity as we stream.
template <bool LAST_USE>
__device__ __forceinline__ void async_tile(uint64_t gbase, uint32_t ldsbase,
                                           uint32_t r4, uint32_t cb, uint32_t rowstride) {
#pragma unroll
  for (int i = 0; i < 8; ++i) {
    uint32_t rr   = (uint32_t)i * 4u + r4;
    uint32_t voff = rr * rowstride + cb;       // byte offset from SGPR base (GVS mode)
    uint32_t ldso = ldsbase + rr * 128u + cb;  // LDS byte address
    if constexpr (LAST_USE) {
      asm volatile("global_load_async_to_lds_b128 %0, %1, %2 th:TH_LOAD_LU"
                   :: "v"(ldso), "v"(voff), "s"(gbase) : "memory");
    } else {
      asm volatile("global_load_async_to_lds_b128 %0, %1, %2"
                   :: "v"(ldso), "v"(voff), "s"(gbase) : "memory");
    }
  }
}

// PASS3=false: compute per-chunk (S, P, Q) transfer coefficients with m_in=v_in=0.
// PASS3=true : re-run the exact scan with (m_in, v_in) from spq[].xy and write outputs.
template <bool PASS3>
__global__ __launch_bounds__(32) void onorm_scan(const float* __restrict__ x,
                                                 float4* __restrict__ spq,
                                                 float* __restrict__ out,
                                                 int F) {
  const int tid   = threadIdx.x;            // 0..31, one feature per lane
  const int fbase = blockIdx.x * BF_BLK;
  const int blk   = blockIdx.y;
  const int row0  = blk * CHUNK;

  __shared__ __align__(16) float smem[2 * TILE * BF_BLK]; // 8 KB, double buffered

  const uint32_t lds0      = (uint32_t)(uintptr_t)(&smem[0]);
  const uint32_t lane      = (uint32_t)tid;
  const uint32_t r4        = lane >> 3;          // 0..3
  const uint32_t cb        = (lane & 7u) << 4;   // 0..112 bytes within row slice
  const uint32_t rowstride = (uint32_t)F * 4u;

  const uint64_t gblk  = (uint64_t)(uintptr_t)(x + (size_t)row0 * F + fbase);
  const uint64_t tstep = (uint64_t)TILE * rowstride;

  const float a  = AFWD;
  const float c1 = 1.0f - AFWD;          // (1-a)
  const float c2 = AFWD * (1.0f - AFWD); // a(1-a)

  float m = 0.0f, v = 0.0f, s = 0.0f, P = 0.0f, Q = 0.0f, g = 1.0f;
  float* outp = nullptr;
  if constexpr (PASS3) {
    float4 mv = spq[(size_t)blk * F + fbase + tid]; // .x = m_in, .y = v_in
    m = mv.x; v = mv.y;
    outp = out + (size_t)row0 * F + fbase + tid;
  }

  // prime the pipeline: two tiles in flight
  async_tile<PASS3>(gblk,         lds0,                  r4, cb, rowstride);
  async_tile<PASS3>(gblk + tstep, lds0 + TILE*BF_BLK*4u, r4, cb, rowstride);

#pragma unroll
  for (int t = 0; t < NTILES; ++t) {
    if (t + 1 < NTILES) wait_async_le8(); else wait_async_le0(); // tile t complete (in-order)
    const float* sb = &smem[(t & 1) * TILE * BF_BLK];

#pragma unroll
    for (int r = 0; r < TILE; ++r) {
      float xv = sb[r * BF_BLK + tid];
      if constexpr (PASS3) {
        float d = xv - m;
        // out is write-once, never re-read: non-temporal store keeps x resident in L2
        __builtin_nontemporal_store(d * rsqrtf(v + EPS_),
                                    &outp[(size_t)(t * TILE + r) * F]);
        v = a * v + c2 * (d * d);
        m = m + c1 * d;
      } else {
        float u = xv - s;        // u_j = x_j - s_{j-1}
        s = s + c1 * u;          // m-scan with m_in = 0
        P = a * P + c2 * (u * u);
        Q = a * Q + c2 * (u * g);
        g = g * a;               // g_j = a^(j-1)
      }
    }

    if (t + 2 < NTILES) {        // refill the buffer we just drained
      wait_ds_le0();
      async_tile<PASS3>(gblk + (uint64_t)(t + 2) * tstep,
                        lds0 + (uint32_t)(t & 1) * (TILE*BF_BLK*4u), r4, cb, rowstride);
    }
  }

  if constexpr (!PASS3) {
    spq[(size_t)blk * F + fbase + tid] = make_float4(s, P, Q, 0.0f);
  }
}

// Sequential combine across chunks (exact): per feature, propagate (m,v) through
// the NB chunk transfer maps, writing each chunk's (m_in, v_in) into spq[].xy,
// and the final m/v into the tail of d_out.
__global__ void onorm_combine(const float* __restrict__ ms, const float* __restrict__ vs,
                              float4* __restrict__ spq, float* __restrict__ out,
                              int F, int NB, unsigned long long BF) {
  int f = blockIdx.x * blockDim.x + threadIdx.x;
  if (f >= F) return;
  float m = ms[f];
  float v = vs[f];

  float aL = AFWD;
#pragma unroll
  for (int i = 0; i < 7; ++i) aL *= aL;          // AFWD^128
  const float Rc = aL * (1.0f - aL);             // closed-form quadratic coefficient

  constexpr int G = 8;                           // group prefetch to hide latency
  for (int b0 = 0; b0 < NB; b0 += G) {
    float4 c[G];
#pragma unroll
    for (int k = 0; k < G; ++k) c[k] = spq[(size_t)(b0 + k) * F + f];
#pragma unroll
    for (int k = 0; k < G; ++k) {
      size_t idx = (size_t)(b0 + k) * F + f;
      *reinterpret_cast<float2*>(&spq[idx]) = make_float2(m, v); // (m_in, v_in) for pass 3
      float vn = aL * v + c[k].y - 2.0f * c[k].z * m + Rc * (m * m); // P - 2Q*m + R*m^2
      float mn = aL * m + c[k].x;                                   // a^L*m + S
      v = vn; m = mn;
    }
  }
  out[BF + f]     = m;   // m_final
  out[BF + F + f] = v;   // v_final
}

extern "C" void kernel_launch(void* const* d_in, const int* in_sizes, int n_in,
                              void* d_out, int out_size, void* d_ws, size_t ws_size,
                              hipStream_t stream) {
  const float* x  = (const float*)d_in[0];
  const float* ms = (const float*)d_in[1];
  const float* vs = (const float*)d_in[2];
  float* out = (float*)d_out;

  const int F  = in_sizes[1];          // 2048
  const int B  = in_sizes[0] / F;      // 16384
  const int NB = B / CHUNK;            // 128

  float4* spq = (float4*)d_ws;         // NB*F float4 = 4 MiB

  dim3 grid(F / BF_BLK, NB);           // 64 x 128 = 8192 wave32 blocks
  onorm_scan<false><<<grid, BF_BLK, 0, stream>>>(x, spq, nullptr, F);
  onorm_combine<<<(F + 255) / 256, 256, 0, stream>>>(ms, vs, spq, out, F, NB,
                                                     (unsigned long long)B * (unsigned long long)F);
  onorm_scan<true><<<grid, BF_BLK, 0, stream>>>(x, spq, out, F);
}